// GCNEncoder_65111704207429
// MI455X (gfx1250) — compile-verified
//
#include <hip/hip_runtime.h>
#include <hip/hip_bf16.h>

#define D 128

typedef float v2f __attribute__((ext_vector_type(2)));
typedef float v8f __attribute__((ext_vector_type(8)));

// ---------------------------------------------------------------------------
// GEMM: H[N,128] = X[N,128] @ W[128,128] + bias, using V_WMMA_F32_16X16X4_F32.
// Block = 256 threads (8 waves). Each wave computes one 16x16 output tile;
// block covers 128 rows x 16 cols. W column-tile staged in LDS as interleaved
// K-pairs so each B fragment is one ds_load_b64, conflict-free.
// ---------------------------------------------------------------------------
__launch_bounds__(256)
__global__ void gemm_bias_wmma(const float* __restrict__ X,
                               const float* __restrict__ W,
                               const float* __restrict__ bias,
                               float* __restrict__ H, int N) {
    // Layout: Ws[((k>>1)*16 + n)*2 + (k&1)] = W[k][tileN+n]
    __shared__ float Ws[128 * 16];
    const int tid   = threadIdx.x;
    const int tileN = blockIdx.y << 4;

#pragma unroll
    for (int i = 0; i < 8; ++i) {
        int idx = tid + (i << 8);        // 0..2047
        int k = idx >> 4, n = idx & 15;
        Ws[(((k >> 1) * 16 + n) << 1) + (k & 1)] = W[k * D + tileN + n];
    }
    __syncthreads();

    const int wave  = tid >> 5;
    const int lane  = tid & 31;
    const int half  = lane >> 4;   // selects K pair (ISA A 16x4 layout)
    const int mrow  = lane & 15;   // M row within tile
    const int tileM = (blockIdx.x << 7) + (wave << 4);

    int rowA = tileM + mrow;
    if (rowA > N - 1) rowA = N - 1;            // clamp: keep EXEC uniform
    const float* xrow = X + (size_t)rowA * D + (half << 1);

    v8f acc = {};
#pragma unroll
    for (int kk = 0; kk < 32; ++kk) {
        v2f a = *(const v2f*)(xrow + (kk << 2));                // A[m][k0+2h..+1]
        int kp = (kk << 1) + half;                              // B rows k0+2h..+1
        v2f b = *(const v2f*)(&Ws[(kp * 16 + mrow) << 1]);
        acc = __builtin_amdgcn_wmma_f32_16x16x4_f32(
            false, a, false, b, (short)0, acc, false, false);
    }

    const int   col = tileN + mrow;
    const float bv  = bias[col];
#pragma unroll
    for (int v = 0; v < 8; ++v) {
        int row = tileM + v + (half << 3);     // C/D layout: VGPR v -> M=v(+8)
        if (row < N) H[(size_t)row * D + col] = acc[v] + bv;
    }
}

// ---------------------------------------------------------------------------
// Utility / graph fill kernels
// ---------------------------------------------------------------------------
__global__ void fill_f32(float* __restrict__ p, long long n, float v) {
    long long i = (long long)blockIdx.x * blockDim.x + threadIdx.x;
    if (i < n) p[i] = v;
}

__global__ void zero4(float4* __restrict__ p, long long n4) {
    long long i = (long long)blockIdx.x * blockDim.x + threadIdx.x;
    if (i < n4) p[i] = make_float4(0.f, 0.f, 0.f, 0.f);
}

__global__ void deg_accum(const int* __restrict__ dst, const float* __restrict__ ew,
                          float* __restrict__ deg, int E) {
    int e = blockIdx.x * 256 + threadIdx.x;
    if (e < E) atomicAdd(&deg[dst[e]], ew[e]);
}

__global__ void rsqrt_inplace(float* __restrict__ p, int n) {
    int i = blockIdx.x * 256 + threadIdx.x;
    if (i < n) p[i] = rsqrtf(p[i]);   // deg >= 1 always (self loop)
}

// src/dst to int32 + precomputed symmetric norm, self-loops appended.
__global__ void build_edges(const int* __restrict__ ei, const float* __restrict__ ew,
                            const float* __restrict__ dinv,
                            int* __restrict__ s32, int* __restrict__ d32,
                            float* __restrict__ nrm, int E, int nE) {
    int i = blockIdx.x * 256 + threadIdx.x;
    if (i >= nE) return;
    int s, d; float w;
    if (i < E) { s = ei[i]; d = ei[E + i]; w = ew[i]; }
    else       { s = d = i - E; w = 1.0f; }
    s32[i] = s; d32[i] = d;
    nrm[i] = dinv[s] * w * dinv[d];
}

// ---------------------------------------------------------------------------
// Edge aggregation: one wave per edge; lane handles 4 features (float4 gather,
// 4x global_atomic_add_f32 scatter). L2-resident.
// ---------------------------------------------------------------------------
__launch_bounds__(256)
__global__ void aggregate(const float* __restrict__ h,
                          const int* __restrict__ s32, const int* __restrict__ d32,
                          const float* __restrict__ nrm,
                          float* __restrict__ agg, int nE) {
    int e = (blockIdx.x << 3) + (threadIdx.x >> 5);
    if (e >= nE) return;
    int lane = threadIdx.x & 31;
    int s = s32[e], d = d32[e];
    float w = nrm[e];
    float4 v = ((const float4*)(h + (size_t)s * D))[lane];
    float* ap = agg + (size_t)d * D + (lane << 2);
    atomicAdd(ap + 0, v.x * w);
    atomicAdd(ap + 1, v.y * w);
    atomicAdd(ap + 2, v.z * w);
    atomicAdd(ap + 3, v.w * w);
}

// ---------------------------------------------------------------------------
// BatchNorm: column sums/sumsq -> scale/shift -> fused apply (+optional ReLU)
// ---------------------------------------------------------------------------
__launch_bounds__(256)
__global__ void bn_stats(const float* __restrict__ x, int N,
                         float* __restrict__ sums, float* __restrict__ sqs) {
    int c = threadIdx.x & 127;
    int r = blockIdx.x * 2 + (threadIdx.x >> 7);
    float s = 0.f, q = 0.f;
    for (; r < N; r += gridDim.x * 2) {
        float v = x[(size_t)r * D + c];
        s += v; q += v * v;
    }
    atomicAdd(&sums[c], s);
    atomicAdd(&sqs[c], q);
}

__global__ void bn_finalize(const float* __restrict__ sums, const float* __restrict__ sqs,
                            const float* __restrict__ g, const float* __restrict__ b,
                            float* __restrict__ sc, float* __restrict__ sh, float invN) {
    int c = threadIdx.x;   // 128 threads
    float mu  = sums[c] * invN;
    float var = fmaxf(sqs[c] * invN - mu * mu, 0.f);
    float inv = rsqrtf(var + 1e-5f);
    float s   = g[c] * inv;
    sc[c] = s;
    sh[c] = b[c] - mu * s;
}

__launch_bounds__(256)
__global__ void bn_apply(const float* __restrict__ in,
                         const float* __restrict__ sc, const float* __restrict__ sh,
                         float* __restrict__ out, long long n4, int relu) {
    long long i = (long long)blockIdx.x * 256 + threadIdx.x;
    if (i >= n4) return;
    int c = (int)(i & 31);                      // 32 float4s per row
    float4 v = ((const float4*)in)[i];
    float4 s = ((const float4*)sc)[c];
    float4 t = ((const float4*)sh)[c];
    float4 r;
    r.x = v.x * s.x + t.x;
    r.y = v.y * s.y + t.y;
    r.z = v.z * s.z + t.z;
    r.w = v.w * s.w + t.w;
    if (relu) {
        r.x = fmaxf(r.x, 0.f); r.y = fmaxf(r.y, 0.f);
        r.z = fmaxf(r.z, 0.f); r.w = fmaxf(r.w, 0.f);
    }
    ((float4*)out)[i] = r;
}

// ---------------------------------------------------------------------------
extern "C" void kernel_launch(void* const* d_in, const int* in_sizes, int n_in,
                              void* d_out, int out_size, void* d_ws, size_t ws_size,
                              hipStream_t stream) {
    const float* x  = (const float*)d_in[0];
    const int*   ei = (const int*)d_in[1];     // [2,E]: row0=src, row1=dst
    const float* ew = (const float*)d_in[2];
    const float* Wl[3]  = {(const float*)d_in[4], (const float*)d_in[6], (const float*)d_in[8]};
    const float* bl[3]  = {(const float*)d_in[5], (const float*)d_in[7], (const float*)d_in[9]};
    const float* gl[3]  = {(const float*)d_in[10], (const float*)d_in[12], (const float*)d_in[14]};
    const float* bel[3] = {(const float*)d_in[11], (const float*)d_in[13], (const float*)d_in[15]};

    const int N  = in_sizes[0] / D;
    const int E  = in_sizes[2];
    const int nE = E + N;

    // Workspace layout (256B aligned slices)
    char*  base = (char*)d_ws;
    size_t off  = 0;
    auto alloc = [&](size_t bytes) -> void* {
        void* p = base + off;
        off += (bytes + 255) & ~(size_t)255;
        return p;
    };
    const size_t nodeBytes = (size_t)N * D * sizeof(float);
    float* bufX  = (float*)alloc(nodeBytes);       // BN output / layer input
    float* bufH  = (float*)alloc(nodeBytes);       // GEMM output
    float* bufA  = (float*)alloc(nodeBytes);       // aggregation output
    int*   s32   = (int*)  alloc((size_t)nE * 4);
    int*   d32   = (int*)  alloc((size_t)nE * 4);
    float* nrm   = (float*)alloc((size_t)nE * 4);
    float* dinv  = (float*)alloc((size_t)N * 4);
    float* stats = (float*)alloc(512 * sizeof(float)); // sums|sqs|scale|shift

    const long long n4 = (long long)N * (D / 4);

    // --- degree / normalization precompute (shared by all 3 layers) ---
    fill_f32<<<(N + 255) / 256, 256, 0, stream>>>(dinv, N, 1.0f);       // self loop
    deg_accum<<<(E + 255) / 256, 256, 0, stream>>>(ei + E, ew, dinv, E);
    rsqrt_inplace<<<(N + 255) / 256, 256, 0, stream>>>(dinv, N);
    build_edges<<<(nE + 255) / 256, 256, 0, stream>>>(ei, ew, dinv, s32, d32, nrm, E, nE);

    // --- three GCN layers ---
    for (int l = 0; l < 3; ++l) {
        const float* Xin = (l == 0) ? x : bufX;
        dim3 ggrid((N + 127) / 128, D / 16);
        gemm_bias_wmma<<<ggrid, 256, 0, stream>>>(Xin, Wl[l], bl[l], bufH, N);

        zero4<<<(int)((n4 + 255) / 256), 256, 0, stream>>>((float4*)bufA, n4);
        aggregate<<<(nE + 7) / 8, 256, 0, stream>>>(bufH, s32, d32, nrm, bufA, nE);

        fill_f32<<<1, 256, 0, stream>>>(stats, 256, 0.0f);
        bn_stats<<<512, 256, 0, stream>>>(bufA, N, stats, stats + 128);
        bn_finalize<<<1, 128, 0, stream>>>(stats, stats + 128, gl[l], bel[l],
                                           stats + 256, stats + 384, 1.0f / (float)N);

        float* Yout = (l == 2) ? (float*)d_out : bufX;
        bn_apply<<<(int)((n4 + 255) / 256), 256, 0, stream>>>(
            bufA, stats + 256, stats + 384, Yout, n4, (l < 2) ? 1 : 0);
    }
}